// KDE_11175504904736
// MI455X (gfx1250) — compile-verified
//
#include <hip/hip_runtime.h>
#include <math.h>

// ---------------------------------------------------------------------------
// Pairwise Gaussian KDE, B=8, N=4096, 2-D points.
//   out[b,i] = norm * sum_j w[b,j] * exp(-(|xi|^2+|xj|^2-2 xi.xj) * inv2s2)
// Compute-bound (134M exps; <0.5MB of I/O). Strategy:
//   * whole batch staged in LDS (48KB) via async global->LDS (CDNA5 path)
//   * 16x16 dot tiles via V_WMMA_F32_16X16X4_F32 (K=2 padded to 4), co-executes
//     with v_exp_f32 on the trans pipe
//   * raw v_exp_f32 (no OCML denorm guard): arg <= 0; underflow->0 is exact
//     for a KDE sum
//   * exp argument computed fully pre-scaled by cexp = -log2(e)/(2*sigma^2):
//       arg[r] = cexp*sqi[r] + cexp*sqj + (-2*cexp)*dot[r],  clamp arg<=0
//     so no per-element rescale multiply is needed before the exp
//   * per-lane accumulate, xor-shuffle reduce over the 16 N-lanes of C/D layout
// ---------------------------------------------------------------------------

#define SIGMA 4.0f
#define N_PTS 4096
#define NBATCH 8
#define WAVES_PER_BLOCK 8
#define TPB (WAVES_PER_BLOCK * 32)
#define BLOCKS_PER_BATCH (N_PTS / (16 * WAVES_PER_BLOCK))  // 32

typedef float v2f __attribute__((ext_vector_type(2)));
typedef float v8f __attribute__((ext_vector_type(8)));
typedef int   v4i_gcc __attribute__((vector_size(16)));  // builtin's pointee type

#if defined(__gfx1250__) && __has_builtin(__builtin_amdgcn_global_load_async_to_lds_b128)
#define HAVE_ASYNC_LDS 1
#else
#define HAVE_ASYNC_LDS 0
#endif

__device__ __forceinline__ float fast_exp2(float x) {
#if defined(__AMDGCN__) && __has_builtin(__builtin_amdgcn_exp2f)
  return __builtin_amdgcn_exp2f(x);  // raw v_exp_f32, no denorm-range fixup
#else
  return exp2f(x);
#endif
}

__global__ __launch_bounds__(TPB) void kde_wmma_kernel(
    const float* __restrict__ w,    // [B, N]
    const float* __restrict__ xy,   // [B, N, 2]
    float* __restrict__ out) {      // [B, N]
  __shared__ float s_xy[2 * N_PTS];  // 32 KB, interleaved (x,y)
  __shared__ float s_w[N_PTS];       // 16 KB

  const int tid  = threadIdx.x;
  const int b    = blockIdx.x / BLOCKS_PER_BATCH;
  const int iblk = blockIdx.x % BLOCKS_PER_BATCH;

  const float* gxy = xy + (size_t)b * (2 * N_PTS);
  const float* gw  = w  + (size_t)b * N_PTS;

#if HAVE_ASYNC_LDS
  // Async DMA global -> LDS, 16B per lane per issue, tracked by ASYNCcnt.
  #pragma unroll
  for (int it = 0; it < 8; ++it) {  // 8192 floats = 2048 float4
    int idx4 = tid + it * TPB;
    __builtin_amdgcn_global_load_async_to_lds_b128(
        (v4i_gcc*)(gxy + idx4 * 4),
        (v4i_gcc*)(&s_xy[idx4 * 4]), 0, 0);
  }
  #pragma unroll
  for (int it = 0; it < 4; ++it) {  // 4096 floats = 1024 float4
    int idx4 = tid + it * TPB;
    __builtin_amdgcn_global_load_async_to_lds_b128(
        (v4i_gcc*)(gw + idx4 * 4),
        (v4i_gcc*)(&s_w[idx4 * 4]), 0, 0);
  }
  asm volatile("s_wait_asynccnt 0" ::: "memory");
#else
  {
    const float4* src = (const float4*)gxy;
    float4* dst = (float4*)s_xy;
    #pragma unroll
    for (int it = 0; it < 8; ++it) dst[tid + it * TPB] = src[tid + it * TPB];
    const float4* srcw = (const float4*)gw;
    float4* dstw = (float4*)s_w;
    #pragma unroll
    for (int it = 0; it < 4; ++it) dstw[tid + it * TPB] = srcw[tid + it * TPB];
  }
#endif
  __syncthreads();

  const int lane = tid & 31;
  const int wv   = tid >> 5;
  const int i0   = (iblk * WAVES_PER_BLOCK + wv) * 16;  // this wave's 16 rows
  const int half = (lane >> 4) & 1;                     // lane half
  const int nIdx = lane & 15;                           // N / column index

  const float inv2s2 = 1.0f / (2.0f * SIGMA * SIGMA);
  const float cexp   = -inv2s2 * 1.4426950408889634f;  // -inv2s2 * log2(e), < 0
  const float c2     = -2.0f * cexp;                   // > 0

  // A matrix (16x4, K=2 padded): lanes 0-15 supply K=0,1 = (x,y) for row M=lane;
  // lanes 16-31 supply K=2,3 = 0.
  const float xi0 = s_xy[2 * (i0 + nIdx) + 0];
  const float yi0 = s_xy[2 * (i0 + nIdx) + 1];
  v2f amat;
  amat.x = half ? 0.0f : xi0;
  amat.y = half ? 0.0f : yi0;

  // cexp * |xi|^2 for the rows this lane's C/D elements touch:
  // VGPR r, lanes 0-15 -> M=r ; lanes 16-31 -> M=r+8.
  float sqi_s[8];
  #pragma unroll
  for (int r = 0; r < 8; ++r) {
    const int m = i0 + r + half * 8;
    const float xx = s_xy[2 * m + 0];
    const float yy = s_xy[2 * m + 1];
    sqi_s[r] = cexp * (xx * xx + yy * yy);
  }

  float acc[8] = {0.f, 0.f, 0.f, 0.f, 0.f, 0.f, 0.f, 0.f};

  for (int j0 = 0; j0 < N_PTS; j0 += 16) {
    const int j = j0 + nIdx;
    const float xj = s_xy[2 * j + 0];
    const float yj = s_xy[2 * j + 1];

    // B matrix (4x16, K=2 padded): lanes 0-15 supply rows K=0,1 for col N=lane;
    // lanes 16-31 supply rows K=2,3 = 0.
    v2f bmat;
    bmat.x = half ? 0.0f : xj;
    bmat.y = half ? 0.0f : yj;

    v8f cz = {};
    // D[m,n] = x_{i0+m} . x_{j0+n}  (exact f32 dot via matrix pipe)
    v8f d = __builtin_amdgcn_wmma_f32_16x16x4_f32(
        /*neg_a=*/false, amat, /*neg_b=*/false, bmat,
        /*c_mod=*/(short)0, cz, /*reuse_a=*/false, /*reuse_b=*/false);

    const float sqj_s = cexp * (xj * xj + yj * yj);
    const float wj    = s_w[j];

    #pragma unroll
    for (int r = 0; r < 8; ++r) {
      // arg = cexp*(sqi + sqj - 2*dot); clamp d2>=0 <=> arg<=0 (cexp<0)
      float arg = (sqi_s[r] + sqj_s) + c2 * d[r];
      arg = fminf(arg, 0.0f);
      acc[r] += wj * fast_exp2(arg);  // native v_exp_f32 on trans pipe
    }
  }

  // Reduce over the 16 column-lanes (xor masks <16 stay within each 16-group).
  #pragma unroll
  for (int r = 0; r < 8; ++r) {
    float v = acc[r];
    v += __shfl_xor(v, 1, 32);
    v += __shfl_xor(v, 2, 32);
    v += __shfl_xor(v, 4, 32);
    v += __shfl_xor(v, 8, 32);
    acc[r] = v;
  }

  const float norm = 1.0f / (2.0f * 3.14159265358979323846f * SIGMA * SIGMA);
  if (nIdx == 0) {
    // lane 0 -> rows i0..i0+7 ; lane 16 -> rows i0+8..i0+15
    float* o = out + (size_t)b * N_PTS + i0 + half * 8;
    #pragma unroll
    for (int r = 0; r < 8; ++r) o[r] = norm * acc[r];
  }
}

extern "C" void kernel_launch(void* const* d_in, const int* in_sizes, int n_in,
                              void* d_out, int out_size, void* d_ws, size_t ws_size,
                              hipStream_t stream) {
  const float* w  = (const float*)d_in[0];   // weights  [B, N]
  const float* xy = (const float*)d_in[1];   // coords   [B, N, 2]
  float* out = (float*)d_out;                // [B, N]
  (void)in_sizes; (void)n_in; (void)out_size; (void)d_ws; (void)ws_size;

  dim3 grid(NBATCH * BLOCKS_PER_BATCH);  // 256 blocks
  dim3 block(TPB);                       // 256 threads = 8 wave32
  kde_wmma_kernel<<<grid, block, 0, stream>>>(w, xy, out);
}